// SpikePool_71648644431983
// MI455X (gfx1250) — compile-verified
//
#include <hip/hip_runtime.h>
#include <hip/hip_bf16.h>
#include <stdint.h>

// SpikePool for MI455X (gfx1250): 2x2x(2-chunk) average pool + dual-phase
// integrate-and-fire. Streaming, memory-bound kernel (320 MiB @ 23.3 TB/s
// => ~14 us floor). Data movement uses the CDNA5 Tensor Data Mover: one
// tensor_load_to_lds DMA per block pulls a contiguous 32 KB tile into LDS
// (TENSORcnt), compute reads via ds_load_b128, results leave via
// non-temporal global_store_b128.
//
// Shapes: x [16,128,64,64,8] f32 -> out [16,128,32,32,8] f32.

#define TH_SPIKE 0.999f
#define USE_TDM 1

typedef float    f32x4 __attribute__((ext_vector_type(4)));
typedef uint32_t u32x4 __attribute__((ext_vector_type(4)));
typedef uint32_t u32x8 __attribute__((ext_vector_type(8)));

// Generic -> LDS(AS3) pointer cast; ptrtoint of an AS3 pointer is the raw
// 32-bit LDS byte offset (what D#.lds_addr / async VDST want).
__device__ __forceinline__ unsigned lds_byte_off(const void* p) {
    return (unsigned)(unsigned long long)(const __attribute__((address_space(3))) char*)p;
}

__global__ __launch_bounds__(256) void spikepool_kernel(const float* __restrict__ x,
                                                        float* __restrict__ out)
{
    constexpr int Wd = 64, Td = 8, H2 = 32, W2 = 32;
    constexpr int ROW_FLOATS  = Wd * Td;           // 512 floats per input h-row
    constexpr int TILE_FLOATS = 16 * ROW_FLOATS;   // 8192 floats = 32 KB

    __shared__ float tile[TILE_FLOATS];

    const int blk = blockIdx.x;
    const int q   = blk & 3;     // which group of 16 input h-rows (8 output rows)
    const int nc  = blk >> 2;    // n*C + c  (0..2047)

    // Contiguous 32 KB input tile for this block.
    const float*   gsrc    = x + (size_t)nc * (64 * ROW_FLOATS) + (size_t)q * TILE_FLOATS;
    const unsigned ldsBase = lds_byte_off(&tile[0]);
    const int      tid     = threadIdx.x;

#if USE_TDM
    // ---- Tensor Data Mover: single-descriptor 1-D DMA, 4096 x 8B elements.
    if (tid < 32) {  // one wave issues the DMA (TDM is per-wave, EXEC-agnostic)
        const uint64_t gaddr = (uint64_t)(uintptr_t)gsrc;

        // D# group 0: [1:0]=count=1 | [63:32]=lds_addr | [120:64]=global_addr
        //             | [127:126]=type=2
        u32x4 g0;
        g0[0] = 1u;
        g0[1] = ldsBase;
        g0[2] = (uint32_t)gaddr;
        g0[3] = ((uint32_t)(gaddr >> 32) & 0x01FFFFFFu) | 0x80000000u;

        // D# group 1:
        //  d0: wg_mask=0 | data_size([17:16])=3 (8B) | no barrier/iterate/pad
        //  d1: atomic_barrier_addr=0 | tensor_dim0[15:0]<<16   (4096 -> 0x1000)
        //  d2: tensor_dim0[31:16]=0  | tensor_dim1[15:0]<<16 = 1
        //  d3: tensor_dim1[31:16]=0  | tile_dim0<<16           (4096)
        //  d4: tile_dim1=0 (unused, 1-D) | tile_dim2=0
        //  d5: tensor_dim0_stride[31:0] = 4096
        //  d6: tensor_dim0_stride[47:32]=0 | tensor_dim1_stride[15:0]=0
        //  d7: tensor_dim1_stride[47:16] = 0
        u32x8 g1;
        g1[0] = 0x00030000u;
        g1[1] = 0x10000000u;
        g1[2] = 0x00010000u;
        g1[3] = 0x10000000u;
        g1[4] = 0u;
        g1[5] = 0x00001000u;
        g1[6] = 0u;
        g1[7] = 0u;

        // 2-group form: VADDR2/VADDR3 = NULL (tensor <= 2D).
        asm volatile("tensor_load_to_lds %0, %1" :: "s"(g0), "s"(g1) : "memory");
#if __has_builtin(__builtin_amdgcn_s_wait_tensorcnt)
        __builtin_amdgcn_s_wait_tensorcnt(0);
#else
        asm volatile("s_wait_tensorcnt 0" ::: "memory");
#endif
    }
    __syncthreads();
#else
    // ---- fallback: per-lane async copy, 2048 x 16B transfers, 8 per thread
    #pragma unroll
    for (int k = 0; k < 8; ++k) {
        const int chunk = tid + (k << 8);                             // 0..2047
        unsigned ldsOff = ldsBase + (unsigned)(chunk << 4);           // *16 B
        uint64_t gaddr  = (uint64_t)(uintptr_t)(gsrc + (chunk << 2)); // *4 floats
        asm volatile("global_load_async_to_lds_b128 %0, %1, off"
                     :: "v"(ldsOff), "v"(gaddr) : "memory");
    }
#if __has_builtin(__builtin_amdgcn_s_wait_asynccnt)
    __builtin_amdgcn_s_wait_asynccnt(0);
#else
    asm volatile("s_wait_asynccnt 0" ::: "memory");
#endif
    __syncthreads();
#endif

    // ---- compute: one output pixel per thread
    const int h2l = tid >> 5;    // 0..7 local output row
    const int w2  = tid & 31;    // 0..31 output column

    // Each row segment: 16 consecutive floats = [w=2w2, t0..7][w=2w2+1, t0..7]
    const float* rowA = &tile[(((2 * h2l)     * Wd) + 2 * w2) * Td];
    const float* rowB = &tile[(((2 * h2l + 1) * Wd) + 2 * w2) * Td];

    f32x4 a0 = *(const f32x4*)(rowA);       // w even, t0..3
    f32x4 a1 = *(const f32x4*)(rowA + 4);   // w even, t4..7
    f32x4 a2 = *(const f32x4*)(rowA + 8);   // w odd,  t0..3
    f32x4 a3 = *(const f32x4*)(rowA + 12);  // w odd,  t4..7
    f32x4 b0 = *(const f32x4*)(rowB);
    f32x4 b1 = *(const f32x4*)(rowB + 4);
    f32x4 b2 = *(const f32x4*)(rowB + 8);
    f32x4 b3 = *(const f32x4*)(rowB + 12);

    // x_out[t] = mean of 8 contributors (4 spatial x 2 time-chunks); in [0,1)
    f32x4 xov = (a0 + a1 + a2 + a3 + b0 + b1 + b2 + b3) * 0.125f;

    float o[8];
    float v0 = 0.f, v1 = 0.f;
    #pragma unroll
    for (int t = 0; t < 4; ++t) {
        const float xo = xov[t];

        // phase 0: input is x_out itself
        float u0 = v0 + xo;
        float s0 = (u0 >= TH_SPIKE) ? 1.f : 0.f;
        v0 = u0 - s0 * TH_SPIKE;
        o[t] = s0;

        // phase 1: input is (x_out % TH) + x_out.  Since x_out in [0, 2*TH),
        // the mod is exactly one conditional subtract (replaces libm fmodf).
        float m  = (xo >= TH_SPIKE) ? (xo - TH_SPIKE) : xo;
        float u1 = v1 + (m + xo);
        float s1 = (u1 >= TH_SPIKE) ? 1.f : 0.f;
        v1 = u1 - s1 * TH_SPIKE;
        o[4 + t] = s1;
    }

    // ---- store: 32 B per thread, contiguous across the block, non-temporal
    const size_t P = (size_t)nc * (H2 * W2) + (size_t)(q * 8 + h2l) * W2 + w2;
    f32x4 lo = {o[0], o[1], o[2], o[3]};
    f32x4 hi = {o[4], o[5], o[6], o[7]};
    f32x4* dst = (f32x4*)(out + P * 8);
    __builtin_nontemporal_store(lo, dst);
    __builtin_nontemporal_store(hi, dst + 1);
}

extern "C" void kernel_launch(void* const* d_in, const int* in_sizes, int n_in,
                              void* d_out, int out_size, void* d_ws, size_t ws_size,
                              hipStream_t stream) {
    const float* x = (const float*)d_in[0];
    float* out = (float*)d_out;
    // 16*128 (n,c) slabs * 4 blocks each = 8192 blocks of 256 threads.
    spikepool_kernel<<<dim3(16 * 128 * 4), dim3(256), 0, stream>>>(x, out);
}